// G2_5858335391841
// MI455X (gfx1250) — compile-verified
//
#include <hip/hip_runtime.h>
#include <math.h>

#define NN 10000   // nodes
#define NE 160000  // edges
#define DD 256     // feature dim

typedef __attribute__((ext_vector_type(2))) float v2f;
typedef __attribute__((ext_vector_type(8))) float v8f;

// ---------------------------------------------------------------------------
// Pass 1: one wave (32 lanes) per edge. Lane L handles floats 4L..4L+3 and 4L+128..4L+131
// via two float4 loads (fully coalesced, X row = 1KB = 8 cachelines).
// deg and neigh_sum accumulated with global_atomic_add_f32 (L2-resident).
// ---------------------------------------------------------------------------
__global__ void scatter_neigh_kernel(const float* __restrict__ X,
                                     const int*   __restrict__ ei,
                                     float*       __restrict__ deg,
                                     float*       __restrict__ nsum) {
  int wave = (blockIdx.x * blockDim.x + threadIdx.x) >> 5;
  int lane = threadIdx.x & 31;
  if (wave >= NE) return;
  int r = ei[wave];        // row (target)
  int c = ei[NE + wave];   // col (source)
  if (lane == 0) atomicAdd(&deg[r], 1.0f);
  const float4* xs = (const float4*)(X + (size_t)c * DD);
  float* dst = nsum + (size_t)r * DD;
#pragma unroll
  for (int j = 0; j < 2; ++j) {
    int q = lane + 32 * j;           // float4 index, 0..63
    float4 v = xs[q];
    int base = q * 4;
    atomicAdd(dst + base + 0, v.x);
    atomicAdd(dst + base + 1, v.y);
    atomicAdd(dst + base + 2, v.z);
    atomicAdd(dst + base + 3, v.w);
  }
}

// ---------------------------------------------------------------------------
// Pass 2: H = relu(X @ Wr + (nsum/deg) @ Wn + b), fp32 WMMA 16x16x4.
// One wave computes one 16x16 tile of H. 625*16 = 10000 tiles.
// A layout (16x4 f32): lanes 0-15 M=lane K={0,1}; lanes 16-31 M=lane-16 K={2,3}.
// B layout (4x16 f32): VGPR0/1 hold rows K={0,1} (lanes 0-15) / K={2,3} (16-31).
// C/D (16x16 f32, v8f): M = v + 8*(lane>=16), N = lane&15.
// ---------------------------------------------------------------------------
__global__ void sage_gemm_kernel(const float* __restrict__ X,
                                 const float* __restrict__ nsum,
                                 const float* __restrict__ deg,
                                 const float* __restrict__ Wr,
                                 const float* __restrict__ Wn,
                                 const float* __restrict__ bias,
                                 float*       __restrict__ H) {
  int tile = blockIdx.x * (blockDim.x >> 5) + (threadIdx.x >> 5);
  int rt = tile >> 4;          // row-tile 0..624
  int ct = tile & 15;          // col-tile 0..15
  int lane = threadIdx.x & 31;
  int lo = lane & 15;
  int hi = lane >> 4;

  int r   = rt * 16 + lo;      // A-matrix row handled by this lane
  int col = ct * 16 + lo;      // B/C/D column handled by this lane

  float invd = 1.0f / fmaxf(deg[r], 1.0f);
  const float* xrow = X    + (size_t)r * DD;
  const float* nrow = nsum + (size_t)r * DD;

  v8f acc = {0.f, 0.f, 0.f, 0.f, 0.f, 0.f, 0.f, 0.f};

  for (int k0 = 0; k0 < DD; k0 += 4) {
    int kk = k0 + 2 * hi;
    // A operands: contiguous pair of K values for this lane's row
    v2f a1 = *(const v2f*)(xrow + kk);
    v2f a2 = (*(const v2f*)(nrow + kk)) * invd;   // fused mean divide
    // B operands: rows kk, kk+1 of the weight matrices at this lane's column
    v2f b1, b2;
    b1.x = Wr[(size_t)kk * DD + col];
    b1.y = Wr[(size_t)(kk + 1) * DD + col];
    b2.x = Wn[(size_t)kk * DD + col];
    b2.y = Wn[(size_t)(kk + 1) * DD + col];
    acc = __builtin_amdgcn_wmma_f32_16x16x4_f32(false, a1, false, b1,
                                                (short)0, acc, false, false);
    acc = __builtin_amdgcn_wmma_f32_16x16x4_f32(false, a2, false, b2,
                                                (short)0, acc, false, false);
  }

  float bv = bias[col];
#pragma unroll
  for (int v = 0; v < 8; ++v) {
    int m = rt * 16 + v + 8 * hi;
    float val = acc[v] + bv;
    H[(size_t)m * DD + col] = fmaxf(val, 0.0f);
  }
}

// ---------------------------------------------------------------------------
// Pass 3: per-edge squared-diff scatter: gg[row] += (H[row]-H[col])^2
// (P == 2.0, so |x|^p == x*x). Wave-per-edge, float4 lanes.
// ---------------------------------------------------------------------------
__global__ void edge_diff_scatter_kernel(const float* __restrict__ H,
                                         const int*   __restrict__ ei,
                                         float*       __restrict__ gg) {
  int wave = (blockIdx.x * blockDim.x + threadIdx.x) >> 5;
  int lane = threadIdx.x & 31;
  if (wave >= NE) return;
  int r = ei[wave];
  int c = ei[NE + wave];
  const float4* hr = (const float4*)(H + (size_t)r * DD);
  const float4* hc = (const float4*)(H + (size_t)c * DD);
  float* dst = gg + (size_t)r * DD;
#pragma unroll
  for (int j = 0; j < 2; ++j) {
    int q = lane + 32 * j;
    float4 a = hr[q];
    float4 b = hc[q];
    float dx = a.x - b.x, dy = a.y - b.y, dz = a.z - b.z, dw = a.w - b.w;
    int base = q * 4;
    atomicAdd(dst + base + 0, dx * dx);
    atomicAdd(dst + base + 1, dy * dy);
    atomicAdd(dst + base + 2, dz * dz);
    atomicAdd(dst + base + 3, dw * dw);
  }
}

// ---------------------------------------------------------------------------
// Pass 4: out = tanh(gg / max(deg,1))
// ---------------------------------------------------------------------------
__global__ void finalize_kernel(const float* __restrict__ gg,
                                const float* __restrict__ deg,
                                float*       __restrict__ out) {
  int i = blockIdx.x * blockDim.x + threadIdx.x;
  if (i >= NN * DD) return;
  int row = i >> 8;  // DD == 256
  float invd = 1.0f / fmaxf(deg[row], 1.0f);
  out[i] = tanhf(gg[i] * invd);
}

extern "C" void kernel_launch(void* const* d_in, const int* in_sizes, int n_in,
                              void* d_out, int out_size, void* d_ws, size_t ws_size,
                              hipStream_t stream) {
  const float* X    = (const float*)d_in[0];
  const int*   ei   = (const int*)  d_in[1];   // [2, E] flat: row then col
  const float* Wr   = (const float*)d_in[2];
  const float* Wn   = (const float*)d_in[3];
  const float* bias = (const float*)d_in[4];
  float* out = (float*)d_out;

  // workspace layout (floats): deg | nsum | gg | H   (~31 MB total)
  float* ws   = (float*)d_ws;
  float* deg  = ws;
  float* nsum = deg + NN;
  float* gg   = nsum + (size_t)NN * DD;
  float* H    = gg   + (size_t)NN * DD;

  // zero the accumulated buffers (deg, nsum, gg are contiguous)
  size_t zero_floats = (size_t)NN + 2 * (size_t)NN * DD;
  hipMemsetAsync(ws, 0, zero_floats * sizeof(float), stream);

  // pass 1: degree + neighbor-sum scatter (8 waves / 256-thread block)
  scatter_neigh_kernel<<<NE / 8, 256, 0, stream>>>(X, ei, deg, nsum);

  // pass 2: fused SAGE GEMM with fp32 WMMA; 10000 tiles, 8 waves/block
  sage_gemm_kernel<<<(NN / 16) * (DD / 16) / 8, 256, 0, stream>>>(
      X, nsum, deg, Wr, Wn, bias, H);

  // pass 3: squared-diff scatter
  edge_diff_scatter_kernel<<<NE / 8, 256, 0, stream>>>(H, ei, gg);

  // pass 4: scatter-mean + tanh
  finalize_kernel<<<(NN * DD) / 256, 256, 0, stream>>>(gg, deg, out);
}